// GauntTensorProduct_17145509446002
// MI455X (gfx1250) — compile-verified
//
#include <hip/hip_runtime.h>
#include <math.h>

// Static problem configuration (matches reference)
#define L1C   2
#define L2C   2
#define LZC   4
#define RB    20
#define RA    19
#define MUL   128
#define CH    128
#define NB    1024

typedef __attribute__((ext_vector_type(2))) float v2f;
typedef __attribute__((ext_vector_type(8))) float v8f;
typedef __attribute__((ext_vector_type(4))) unsigned int v4u;
typedef __attribute__((ext_vector_type(4))) int v4i;
typedef __attribute__((ext_vector_type(8))) int v8i;

// ---------------------------------------------------------------------------
// Kernel 0: compute Gaunt coefficient tensor G[25][9][9] on device.
//   - Gauss-Legendre nodes/weights (RB=20) via Newton iteration in f64
//   - Real orthonormal spherical harmonics Y[lm][b][a] on the grid (in LDS)
//   - G[d',d1,d2] = (2*pi/RA) * sum_{b,a} qw[b] * Y[d'] * Y[d1] * Y[d2]
// This makes the synthesis->product->analysis chain an exact bilinear form.
// ---------------------------------------------------------------------------
__global__ __launch_bounds__(512) void setup_tables_kernel(float* __restrict__ G)
{
    __shared__ float sct[RB];
    __shared__ float sqw[RB];
    __shared__ float sY[25 * RB * RA];   // 38 KB
    const double PI = 3.14159265358979323846;

    // Phase 1: Gauss-Legendre nodes of P_20 (one thread per node)
    if (threadIdx.x < RB) {
        int i = threadIdx.x;
        double x = cos(PI * (i + 0.75) / (RB + 0.5));
        double dp = 1.0;
        for (int it = 0; it < 100; ++it) {
            double p0 = 1.0, p1 = x;
            for (int k = 2; k <= RB; ++k) {
                double p2 = ((2.0 * k - 1.0) * x * p1 - (k - 1.0) * p0) / k;
                p0 = p1; p1 = p2;
            }
            dp = RB * (x * p1 - p0) / (x * x - 1.0);
            x -= p1 / dp;
        }
        sct[i] = (float)x;
        sqw[i] = (float)(2.0 / ((1.0 - x * x) * dp * dp));
    }
    __syncthreads();

    // Phase 2: Y table on the (beta, alpha) grid
    for (int e = threadIdx.x; e < 25 * RB * RA; e += blockDim.x) {
        int lm  = e / (RB * RA);
        int rem = e % (RB * RA);
        int b = rem / RA;
        int a = rem % RA;
        int l = 0;
        while ((l + 1) * (l + 1) <= lm) ++l;       // l = floor(sqrt(lm))
        int m  = lm - l * l - l;
        int am = m < 0 ? -m : m;

        double fac = 1.0;                           // (l+|m|)! / (l-|m|)!
        for (int t = l - am + 1; t <= l + am; ++t) fac *= (double)t;
        double nlm = sqrt((2.0 * l + 1.0) / (4.0 * PI) / fac);

        double xb = (double)sct[b];
        // Associated Legendre P_l^{am}(xb), Condon-Shortley phase
        double pmm = 1.0;
        if (am > 0) {
            double df = 1.0;
            for (int k = 1; k < 2 * am; k += 2) df *= (double)k;  // (2m-1)!!
            double s = sqrt(1.0 - xb * xb);
            double sp = 1.0;
            for (int k = 0; k < am; ++k) sp *= s;
            pmm = ((am & 1) ? -1.0 : 1.0) * df * sp;
        }
        double P;
        if (l == am) {
            P = pmm;
        } else {
            double pmmp1 = xb * (2.0 * am + 1.0) * pmm;
            if (l == am + 1) {
                P = pmmp1;
            } else {
                for (int ll = am + 2; ll <= l; ++ll) {
                    double p = ((2.0 * ll - 1.0) * xb * pmmp1 - (ll + am - 1.0) * pmm) / (double)(ll - am);
                    pmm = pmmp1; pmmp1 = p;
                }
                P = pmmp1;
            }
        }
        double alpha = 2.0 * PI * (double)a / (double)RA;
        double ang;
        if (m == 0)      ang = 1.0;
        else if (m > 0)  ang = sqrt(2.0) * cos((double)m * alpha);
        else             ang = sqrt(2.0) * sin((double)am * alpha);
        sY[e] = (float)(nlm * P * ang);
    }
    __syncthreads();

    // Phase 3: Gaunt tensor G[25][9][9]
    const float norm = (float)(2.0 * PI / (double)RA);
    for (int e = threadIdx.x; e < 25 * 9 * 9; e += blockDim.x) {
        int dp = e / 81;
        int d1 = (e / 9) % 9;
        int d2 = e % 9;
        const float* Yp = &sY[dp * RB * RA];
        const float* Y1 = &sY[d1 * RB * RA];
        const float* Y2 = &sY[d2 * RB * RA];
        float s = 0.f;
        for (int b = 0; b < RB; ++b) {
            float qb = sqw[b];
            float sb = 0.f;
            for (int a = 0; a < RA; ++a) {
                int g = b * RA + a;
                sb = fmaf(Yp[g] * Y1[g], Y2[g], sb);
            }
            s = fmaf(qb, sb, s);
        }
        G[e] = s * norm;
    }
}

// ---------------------------------------------------------------------------
// Generic per-l equivariant linear as a WMMA f32 GEMM with TDM-staged B.
//   A logical [Nrows x K], Nrows = NB*DEG, A[(n,j)][k] = Abase[n*K*DIN + k*DIN + D0+j]
//   B = W [K x CD], staged into LDS once per block by the Tensor Data Mover
//   Out[(n,j)][c]  ->  O[n*CD*DOUT + c*DOUT + D0+j], scaled.
// One wave owns a 16x128 tile (8 x v8f acc). 8 waves/block = 128 rows/block.
// Uses v_wmma_f32_16x16x4_f32 (full f32, K folded in steps of 4).
// Fragment layouts per CDNA5 ISA 7.12.2:
//   A 16x4: lanes 0-15 hold K=0,1 ; lanes 16-31 hold K=2,3 (M = lane&15)
//   B 4x16: mirrored, lane&15 = column
//   C/D: VGPR r -> M = r + 8*(lane>=16), N = lane&15
// All tile factors are compile-time constants: DEG-divisions fold to
// multiply-shift, loops have static trip counts (no zero-trip guards).
// ---------------------------------------------------------------------------
template <int K, int CD, int DIN, int DOUT, int DEG>
__global__ __launch_bounds__(256) void lin_gemm_wmma(
    const float* __restrict__ A,   // [NB][K][DIN]
    const float* __restrict__ W,   // [K][CD]
    float* __restrict__ O,         // [NB][CD][DOUT]
    float scale)
{
    constexpr int LL = (DEG - 1) / 2;
    constexpr int D0 = LL * LL;

    extern __shared__ float sW[];  // K*CD floats (64 KB), TDM destination

    // ---- Stage W -> LDS via Tensor Data Mover (wave 0 issues; EXEC ignored
    //      by tensor ops). D# per CDNA5 ISA:
    //      2D tensor, data_size=4B, tile = whole [K][CD] matrix.
    if (threadIdx.x < 32) {
        unsigned long long ga = (unsigned long long)(uintptr_t)W;
        // flat->LDS: addr[31:0] is the LDS byte offset (aperture rule)
        unsigned int lds_off = (unsigned int)(uintptr_t)&sW[0];

        v4u g0;
        g0[0] = 1u;                                        // count=1, user mode
        g0[1] = lds_off;                                   // lds_addr (bytes)
        g0[2] = (unsigned int)(ga & 0xFFFFFFFFu);          // global_addr[31:0]
        g0[3] = (unsigned int)((ga >> 32) & 0x01FFFFFFu)   // global_addr[56:32]
              | (2u << 30);                                // type=2 ("image")

        v8i g1;
        g1[0] = 0x00020000;                  // wg_mask=0, data_size=2 (4B)
        g1[1] = (int)((unsigned)CD << 16);   // bits 63:48  = tensor_dim0[15:0]
        g1[2] = (int)((unsigned)K  << 16);   // bits 95:80  = tensor_dim1[15:0]
        g1[3] = (int)((unsigned)CD << 16);   // bits 127:112 = tile_dim0
        g1[4] = K;                           // bits 143:128 = tile_dim1 (tile_dim2=0)
        g1[5] = CD;                          // bits 191:160 = tensor_dim0_stride lo
        g1[6] = 0;                           // stride hi / dim1_stride lo (unused, 2D)
        g1[7] = 0;

        v4i gz4 = {0, 0, 0, 0};              // groups 2/3 unused (<=2D tensor)
        v8i gz8 = {0, 0, 0, 0, 0, 0, 0, 0};  // extra group (6-arg toolchain form)
        __builtin_amdgcn_tensor_load_to_lds(g0, g1, gz4, gz4, gz8, 0);
        __builtin_amdgcn_s_wait_tensorcnt(0);
    }
    __syncthreads();

    const int lane  = threadIdx.x & 31;
    const int wave  = threadIdx.x >> 5;
    const int M     = lane & 15;
    const int kslot = lane >> 4;                       // 0 or 1
    const int rowbase = blockIdx.x * 128 + wave * 16;  // wave's 16-row tile
    const int row = rowbase + M;
    const int n = row / DEG;                           // constant divisor
    const int j = row - n * DEG;
    const float* __restrict__ Ar = A + (size_t)n * K * DIN + (D0 + j);

    const v8f zerov = {0.f, 0.f, 0.f, 0.f, 0.f, 0.f, 0.f, 0.f};
    v8f acc[8];
#pragma unroll
    for (int t = 0; t < 8; ++t) acc[t] = zerov;

#pragma unroll 4
    for (int kb = 0; kb < K; kb += 4) {
        if ((kb & 15) == 0)  // statically resolved after unroll-by-4
            __builtin_prefetch(Ar + (size_t)(kb + 16) * DIN, 0, 3);
        const int k0 = kb + kslot * 2;
        v2f a;
        a.x = Ar[(size_t)k0 * DIN];
        a.y = Ar[(size_t)(k0 + 1) * DIN];
        const float* __restrict__ W0 = &sW[(size_t)k0 * CD + M];
#pragma unroll
        for (int ct = 0; ct < 8; ++ct) {
            v2f b;
            b.x = W0[ct * 16];        // ds_load from staged W
            b.y = W0[CD + ct * 16];
            acc[ct] = __builtin_amdgcn_wmma_f32_16x16x4_f32(
                false, a, false, b, (short)0, acc[ct], false, false);
        }
    }

    // Epilogue: scale + scatter to [n][c][DOUT] layout
#pragma unroll
    for (int ct = 0; ct < 8; ++ct) {
        const int col = ct * 16 + M;
#pragma unroll
        for (int r = 0; r < 8; ++r) {
            const int gm   = r + kslot * 8;
            const int grow = rowbase + gm;
            const int gn   = grow / DEG;
            const int gj   = grow - gn * DEG;
            O[(size_t)gn * CD * DOUT + (size_t)col * DOUT + (D0 + gj)] = acc[ct][r] * scale;
        }
    }
}

// ---------------------------------------------------------------------------
// Gaunt bilinear stage: z[n,c,:25] = G : (xc[n,c,:9] outer yc[n,c,:9])
// One thread per (n,c). G broadcast via LDS (8.1 KB).
// ---------------------------------------------------------------------------
__global__ __launch_bounds__(256) void gaunt_bilinear_kernel(
    const float* __restrict__ xc,  // [NB*CH][9]
    const float* __restrict__ yc,  // [NB*CH][9]
    const float* __restrict__ G,   // [25][9][9]
    float* __restrict__ z)         // [NB*CH][25]
{
    __shared__ float sG[25 * 81];
    for (int i = threadIdx.x; i < 25 * 81; i += blockDim.x) sG[i] = G[i];
    __syncthreads();

    const size_t idx = (size_t)blockIdx.x * blockDim.x + threadIdx.x;  // n*CH + c
    const float* xp = xc + idx * 9;
    const float* yp = yc + idx * 9;
    float xv[9], yv[9];
#pragma unroll
    for (int i = 0; i < 9; ++i) { xv[i] = xp[i]; yv[i] = yp[i]; }
    float xy[81];
#pragma unroll
    for (int d1 = 0; d1 < 9; ++d1)
#pragma unroll
        for (int d2 = 0; d2 < 9; ++d2) xy[d1 * 9 + d2] = xv[d1] * yv[d2];

    float* zp = z + idx * 25;
    for (int dp = 0; dp < 25; ++dp) {
        const float* g = &sG[dp * 81];
        float s = 0.f;
#pragma unroll
        for (int q = 0; q < 81; ++q) s = fmaf(g[q], xy[q], s);
        zp[dp] = s;
    }
}

// ---------------------------------------------------------------------------
// Host launcher
// ---------------------------------------------------------------------------
extern "C" void kernel_launch(void* const* d_in, const int* in_sizes, int n_in,
                              void* d_out, int out_size, void* d_ws, size_t ws_size,
                              hipStream_t stream)
{
    const float* x  = (const float*)d_in[0];  // [NB][MUL][9]
    const float* y  = (const float*)d_in[1];  // [NB][MUL][9]
    const float* wx = (const float*)d_in[2];  // [3][MUL][CH]
    const float* wy = (const float*)d_in[3];  // [3][MUL][CH]
    const float* wz = (const float*)d_in[4];  // [5][CH][CH]
    float* out = (float*)d_out;               // [NB][CH][25]

    float* ws = (float*)d_ws;
    float* G  = ws;                                    // 2025 floats (pad to 4096)
    float* xc = ws + 4096;                             // NB*CH*9
    float* yc = xc + (size_t)NB * CH * 9;              // NB*CH*9
    float* z  = yc + (size_t)NB * CH * 9;              // NB*CH*25

    const size_t ldsW = (size_t)MUL * CH * sizeof(float);   // 64 KB staged W

    // 1) Gaunt coefficient tensor (exact, device-computed, capture-safe)
    setup_tables_kernel<<<dim3(1), dim3(512), 0, stream>>>(G);

    // 2) Per-l linear_in for x and y (WMMA f32 GEMMs, K = MUL = 128)
    //    blocks = NB*deg/128 = 8*deg
    const float sc_in = 1.0f / sqrtf((float)MUL);
    lin_gemm_wmma<128, 128, 9, 9, 1><<< 8, 256, ldsW, stream>>>(x, wx + 0 * MUL * CH, xc, sc_in);
    lin_gemm_wmma<128, 128, 9, 9, 3><<<24, 256, ldsW, stream>>>(x, wx + 1 * MUL * CH, xc, sc_in);
    lin_gemm_wmma<128, 128, 9, 9, 5><<<40, 256, ldsW, stream>>>(x, wx + 2 * MUL * CH, xc, sc_in);
    lin_gemm_wmma<128, 128, 9, 9, 1><<< 8, 256, ldsW, stream>>>(y, wy + 0 * MUL * CH, yc, sc_in);
    lin_gemm_wmma<128, 128, 9, 9, 3><<<24, 256, ldsW, stream>>>(y, wy + 1 * MUL * CH, yc, sc_in);
    lin_gemm_wmma<128, 128, 9, 9, 5><<<40, 256, ldsW, stream>>>(y, wy + 2 * MUL * CH, yc, sc_in);

    // 3) Bilinear Gaunt contraction (replaces grid synth/product/analysis exactly)
    gaunt_bilinear_kernel<<<(NB * CH) / 256, 256, 0, stream>>>(xc, yc, G, z);

    // 4) Per-l linear_out (WMMA f32 GEMMs, K = CH = 128)
    const float sc_out = 1.0f / sqrtf((float)CH);
    lin_gemm_wmma<128, 128, 25, 25, 1><<< 8, 256, ldsW, stream>>>(z, wz + 0 * CH * CH, out, sc_out);
    lin_gemm_wmma<128, 128, 25, 25, 3><<<24, 256, ldsW, stream>>>(z, wz + 1 * CH * CH, out, sc_out);
    lin_gemm_wmma<128, 128, 25, 25, 5><<<40, 256, ldsW, stream>>>(z, wz + 2 * CH * CH, out, sc_out);
    lin_gemm_wmma<128, 128, 25, 25, 7><<<56, 256, ldsW, stream>>>(z, wz + 3 * CH * CH, out, sc_out);
    lin_gemm_wmma<128, 128, 25, 25, 9><<<72, 256, ldsW, stream>>>(z, wz + 4 * CH * CH, out, sc_out);
}